// FeaturePropagation_57956288692296
// MI455X (gfx1250) — compile-verified
//
#include <hip/hip_runtime.h>
#include <stdint.h>

#define N_NODES   50000
#define D_FEAT    128
#define N_EDGES   1600000
#define N_ITER    20
#define CHUNK     64          // packed edges staged per async step: 64*8B = one b128/lane wave-wide
#define WAVES_PB  8           // 256-thread block = 8 wave32s, one node per wave

// ---------------------------------------------------------------------------
// One-shot preprocessing kernels (run once per launch; cost << 20 SpMM iters)
// ---------------------------------------------------------------------------

// degree over col (for D^-1/2) and per-row edge counts (for CSR build)
__global__ void count_kernel(const int* __restrict__ row, const int* __restrict__ col,
                             int* __restrict__ degi, int* __restrict__ cnt) {
    int e = blockIdx.x * 256 + threadIdx.x;
    if (e < N_EDGES) {
        atomicAdd(&degi[col[e]], 1);
        atomicAdd(&cnt[row[e]], 1);
    }
}

__global__ void dinv_kernel(const int* __restrict__ degi, float* __restrict__ dis) {
    int i = blockIdx.x * 256 + threadIdx.x;
    if (i < N_NODES) {
        int d = degi[i];
        dis[i] = (d > 0) ? rsqrtf((float)d) : 0.0f;
    }
}

// Single-block exclusive scan of per-row counts, padding each row's segment to a
// multiple of 4 packed entries (32B) so async-LDS staging of CSR segments is aligned.
// Writes row_start[0..N] and re-initializes cur[] as the fill cursor.
__global__ void scan_kernel(int* __restrict__ cur, int* __restrict__ row_start, int n) {
    __shared__ int sm[1024];
    __shared__ int carry;
    int tid = threadIdx.x;
    if (tid == 0) carry = 0;
    __syncthreads();
    for (int base = 0; base < n; base += 1024) {
        int i = base + tid;
        int v = (i < n) ? cur[i] : 0;
        int v4 = (v + 3) & ~3;          // pad row segment to multiple of 4 entries
        sm[tid] = v4;
        __syncthreads();
        for (int off = 1; off < 1024; off <<= 1) {
            int t = (tid >= off) ? sm[tid - off] : 0;
            __syncthreads();
            sm[tid] += t;
            __syncthreads();
        }
        int incl = sm[tid];
        int excl = incl - v4;
        int c = carry;
        if (i < n) {
            row_start[i] = c + excl;
            cur[i]       = c + excl;    // fill cursor
        }
        __syncthreads();
        if (tid == 1023) carry = c + sm[1023];
        __syncthreads();
    }
    if (tid == 0) row_start[n] = carry;
}

// Scatter edges into packed CSR slots {col, w_bits}; padding gaps stay {0, 0.0f}.
__global__ void fill_kernel(const int* __restrict__ row, const int* __restrict__ col,
                            const float* __restrict__ dis, int* __restrict__ cur,
                            int2* __restrict__ csr) {
    int e = blockIdx.x * 256 + threadIdx.x;
    if (e < N_EDGES) {
        int r = row[e], c = col[e];
        int p = atomicAdd(&cur[r], 1);
        float wv = dis[r] * dis[c];
        csr[p] = make_int2(c, __float_as_int(wv));
    }
}

// out0 = mask ? x : 0
__global__ void init_kernel(const float* __restrict__ x, const unsigned char* __restrict__ mask,
                            float* __restrict__ out) {
    int t = blockIdx.x * 256 + threadIdx.x;   // one float4 per thread
    if (t < N_NODES * (D_FEAT / 4)) {
        size_t o = (size_t)t * 4;
        float4 xv = *(const float4*)(x + o);
        uchar4 mv = *(const uchar4*)(mask + o);
        float4 r;
        r.x = mv.x ? xv.x : 0.0f;
        r.y = mv.y ? xv.y : 0.0f;
        r.z = mv.z ? xv.z : 0.0f;
        r.w = mv.w ? xv.w : 0.0f;
        *(float4*)(out + o) = r;
    }
}

// ---------------------------------------------------------------------------
// Hot kernel: one wave32 per destination node. The node's packed CSR segment
// ({col,w} pairs) is staged into LDS with one CDNA5 async global->LDS b128 per
// 64 edges (ASYNCcnt), then each edge costs one uniform ds_load_b64 broadcast
// + one float4 gather of the (L2-resident) source row + 4 FMAs per lane.
// No atomics anywhere in the 20-iteration loop.
// ---------------------------------------------------------------------------
__global__ void __launch_bounds__(256)
prop_kernel(const float* __restrict__ src, float* __restrict__ dst,
            const float* __restrict__ x, const unsigned char* __restrict__ mask,
            const int* __restrict__ row_start, const int2* __restrict__ csr) {
    __shared__ int2 smem[WAVES_PB * CHUNK];      // 512B staging per wave

    const int lane = threadIdx.x & 31;
    const int wv   = threadIdx.x >> 5;
    const int node = blockIdx.x * WAVES_PB + wv;
    if (node >= N_NODES) return;                 // uniform per wave

    const int beg = row_start[node];
    const int end = row_start[node + 1];

    // low 32 bits of the generic (flat) pointer to LDS == LDS byte offset
    const int2* sld = &smem[wv * CHUNK];
    const unsigned lc = (unsigned)(size_t)sld + (unsigned)(lane << 4);   // b128/lane

    float4 acc = make_float4(0.0f, 0.0f, 0.0f, 0.0f);
    const int fo = lane << 2;                    // this lane's 4 features

    for (int j = beg; j < end; j += CHUNK) {
        const int n = min(CHUNK, end - j);
        unsigned long long ga = (unsigned long long)(size_t)(csr + j) + (unsigned)(lane << 4);
        // don't clobber LDS while prior ds reads are in flight
        asm volatile("s_wait_dscnt 0x0" ::: "memory");
        asm volatile("global_load_async_to_lds_b128 %0, %1, off" :: "v"(lc), "v"(ga) : "memory");
        asm volatile("s_wait_asynccnt 0x0" ::: "memory");
#pragma unroll 4
        for (int k = 0; k < n; ++k) {
            const int2  cw = sld[k];                         // uniform ds_load_b64 broadcast
            const float w  = __int_as_float(cw.y);           // 0 in padding gaps
            const float4 s = *(const float4*)(src + (size_t)cw.x * D_FEAT + fo);
            acc.x = fmaf(w, s.x, acc.x);
            acc.y = fmaf(w, s.y, acc.y);
            acc.z = fmaf(w, s.z, acc.z);
            acc.w = fmaf(w, s.w, acc.w);
        }
    }

    const size_t o = (size_t)node * D_FEAT + fo;
    const float4 xv = *(const float4*)(x + o);
    const uchar4 mv = *(const uchar4*)(mask + o);
    float4 r;
    r.x = mv.x ? xv.x : acc.x;
    r.y = mv.y ? xv.y : acc.y;
    r.z = mv.z ? xv.z : acc.z;
    r.w = mv.w ? xv.w : acc.w;
    *(float4*)(dst + o) = r;
}

// ---------------------------------------------------------------------------
// Launch: build packed CSR once, then 20 ping-pong SpMM iterations (final -> d_out)
// ---------------------------------------------------------------------------
extern "C" void kernel_launch(void* const* d_in, const int* in_sizes, int n_in,
                              void* d_out, int out_size, void* d_ws, size_t ws_size,
                              hipStream_t stream) {
    const float*         x    = (const float*)d_in[0];
    const int*           ei   = (const int*)d_in[1];
    const unsigned char* mask = (const unsigned char*)d_in[2];
    const int* row = ei;
    const int* col = ei + N_EDGES;
    float* out = (float*)d_out;

    // workspace carve (all 16B aligned)
    const int CSR_CAP = N_EDGES + 4 * N_NODES + 128;   // padded rows + staging slack
    char* w = (char*)d_ws;
    int*   degi   = (int*)w;   w += (size_t)N_NODES * 4;        // 200000
    int*   cur    = (int*)w;   w += (size_t)N_NODES * 4;        // 200000
    float* dis    = (float*)w; w += (size_t)N_NODES * 4;        // 200000
    int*   rstart = (int*)w;   w += (size_t)(N_NODES + 4) * 4;  // 200016 (16B multiple)
    int2*  csr    = (int2*)w;  w += (size_t)CSR_CAP * 8;        // 14.4 MB
    float* bufB   = (float*)w;                                  // 25.6 MB

    // zero counters + packed CSR (padding gaps must be {0, 0.0f})
    hipMemsetAsync(degi, 0, (size_t)N_NODES * 4 * 2, stream);   // degi + cur contiguous
    hipMemsetAsync(csr, 0, (size_t)CSR_CAP * 8, stream);

    const int eb = (N_EDGES + 255) / 256;
    const int nb = (N_NODES + 255) / 256;

    count_kernel<<<eb, 256, 0, stream>>>(row, col, degi, cur);
    dinv_kernel<<<nb, 256, 0, stream>>>(degi, dis);
    scan_kernel<<<1, 1024, 0, stream>>>(cur, rstart, N_NODES);
    fill_kernel<<<eb, 256, 0, stream>>>(row, col, dis, cur, csr);

    const int ib = (N_NODES * (D_FEAT / 4) + 255) / 256;
    init_kernel<<<ib, 256, 0, stream>>>(x, mask, out);          // out0 -> d_out (P0)

    const int pb = (N_NODES + WAVES_PB - 1) / WAVES_PB;
    float* P0 = out;   // even-iteration results land here; iter 20 (even) -> d_out
    float* P1 = bufB;
    for (int t = 0; t < N_ITER; ++t) {
        const float* s = (t & 1) ? P1 : P0;
        float*       d = (t & 1) ? P0 : P1;
        prop_kernel<<<pb, 256, 0, stream>>>(s, d, x, mask, rstart, csr);
    }
    // N_ITER = 20 (even) => final result is in P0 == d_out
}